// CTCLayer_55336358642048
// MI455X (gfx1250) — compile-verified
//
#include <hip/hip_runtime.h>

// CTC batch cost, B=256, T=512, C=512, L=64  ->  S=129, blank=511
#define CTC_B 256
#define CTC_T 512
#define CTC_C 512
#define CTC_L 64
#define CTC_S (2 * CTC_L + 1) /* 129 */
#define BLANK (CTC_C - 1)
#define NEGV (-1e30f)
#define EPSV (1e-7f)

#define NTHREADS 160 /* 5 waves: lanes 0..128 = states, 0..127 = copy engines */
#define NCOPY 128    /* 128 lanes * 16B = one 2KB row per async issue group   */
#define DEPTH 8      /* async row-pipeline depth (power of two)               */

typedef int v4i __attribute__((ext_vector_type(4)));

// ---- CDNA5 async global->LDS copy + asynccnt wait ------------------------
#if defined(__HIP_DEVICE_COMPILE__)

#if __has_builtin(__builtin_amdgcn_global_load_async_to_lds_b128)
__device__ __forceinline__ void async_cp16(const float* g, float* l) {
  __builtin_amdgcn_global_load_async_to_lds_b128(
      (__attribute__((address_space(1))) v4i*)g,
      (__attribute__((address_space(3))) v4i*)l, 0, 0);
}
#else
__device__ __forceinline__ void async_cp16(const float* g, float* l) {
  unsigned loff =
      (unsigned)(unsigned long long)(__attribute__((address_space(3))) void*)l;
  asm volatile("global_load_async_to_lds_b128 %0, %1, off"
               :
               : "v"(loff), "v"(g)
               : "memory");
}
#endif

#if __has_builtin(__builtin_amdgcn_s_wait_asynccnt)
#define WAIT_ASYNC(n) __builtin_amdgcn_s_wait_asynccnt(n)
#else
#define WAIT_ASYNC(n) asm volatile("s_wait_asynccnt %0" ::"n"(n) : "memory")
#endif

#else // host pass stubs (never executed)
__device__ inline void async_cp16(const float*, float*) {}
#define WAIT_ASYNC(n)
#endif

__global__ __launch_bounds__(NTHREADS) void ctc_alpha_kernel(
    const int* __restrict__ y_true,   // [B, L]
    const float* __restrict__ y_pred, // [B, T, C] probabilities
    float* __restrict__ out)          // [B] (== [B,1] flat)
{
  __shared__ float rows[DEPTH][CTC_C]; // deep async row pipeline (16 KB)
  __shared__ float alphaBuf[2][CTC_S]; // ping-pong alpha
  __shared__ int extS[CTC_S];          // extended label sequence
  __shared__ unsigned char skipS[CTC_S];

  const int b = blockIdx.x;
  const int tid = threadIdx.x;
  const float* base = y_pred + (size_t)b * CTC_T * CTC_C;

  // Build extended labels: even -> blank, odd -> y_true[b, s/2]
  if (tid < CTC_S) {
    int e = BLANK;
    if (tid & 1) e = y_true[b * CTC_L + (tid >> 1)];
    extS[tid] = e;
  }

  // Prologue: fill the pipeline with rows 0..DEPTH-1 (ASYNCcnt = DEPTH)
  if (tid < NCOPY) {
#pragma unroll
    for (int r = 0; r < DEPTH; ++r)
      async_cp16(base + (size_t)r * CTC_C + tid * 4, &rows[r][tid * 4]);
  }
  __syncthreads(); // extS visible

  // skip_ok[s] = (ext[s] != blank) && (ext[s] != ext[s-2]); only used for s>=2
  if (tid < CTC_S) {
    int ok = 0;
    if (tid & 1) ok = (tid < 3) ? 1 : (extS[tid] != extS[tid - 2]);
    skipS[tid] = (unsigned char)ok;
  }

  WAIT_ASYNC(DEPTH - 1); // row 0 landed (copy waves; others trivially pass)
  __syncthreads();

  float* pa = alphaBuf[0];
  float* pb = alphaBuf[1];

  // alpha0: states 0,1 get lp(t=0), rest NEG
  if (tid < CTC_S) {
    float v = NEGV;
    if (tid < 2) v = __logf(rows[0][extS[tid]] + EPSV);
    pa[tid] = v;
  }
  __syncthreads(); // rows[0] fully consumed, alpha0 visible

  // Refill slot 0 with row DEPTH (pipeline depth restored to DEPTH)
  if (tid < NCOPY)
    async_cp16(base + (size_t)DEPTH * CTC_C + tid * 4, &rows[0][tid * 4]);

  for (int t = 1; t < CTC_T; ++t) {
    WAIT_ASYNC(DEPTH - 1); // oldest outstanding row (== row t) has arrived
    __syncthreads();

    const float* row = rows[t & (DEPTH - 1)];
    if (tid < CTC_S) {
      float a0 = pa[tid];
      float a1 = (tid >= 1) ? pa[tid - 1] : NEGV;
      float a2 = (tid >= 2 && skipS[tid]) ? pa[tid - 2] : NEGV;
      float m = fmaxf(a0, fmaxf(a1, a2));
      float s = __expf(a0 - m) + __expf(a1 - m) + __expf(a2 - m);
      float lp = __logf(row[extS[tid]] + EPSV);
      pb[tid] = m + __logf(s) + lp;
    }
    __syncthreads(); // rows[t & (DEPTH-1)] consumed, pb complete

    // Prefetch row t+DEPTH (clamped so WAIT_ASYNC stays a constant) into the
    // buffer just consumed. Clamped duplicates only land in buffers that are
    // never read again before their own post-barrier overwrite -> race-free.
    int nr = t + DEPTH;
    if (nr > CTC_T - 1) nr = CTC_T - 1;
    if (tid < NCOPY)
      async_cp16(base + (size_t)nr * CTC_C + tid * 4,
                 &rows[t & (DEPTH - 1)][tid * 4]);

    float* tmp = pa;
    pa = pb;
    pb = tmp;
  }

  // -ll = -(LSE(alpha[S-1], alpha[S-2]))
  if (tid == 0) {
    float a = pa[CTC_S - 1];
    float c = pa[CTC_S - 2];
    float m = fmaxf(a, c);
    float ll = m + __logf(__expf(a - m) + __expf(c - m));
    out[b] = -ll;
  }
}

extern "C" void kernel_launch(void* const* d_in, const int* in_sizes, int n_in,
                              void* d_out, int out_size, void* d_ws,
                              size_t ws_size, hipStream_t stream) {
  (void)in_sizes;
  (void)n_in;
  (void)out_size;
  (void)d_ws;
  (void)ws_size;
  const int* y_true = (const int*)d_in[0];     // [256, 64] int32
  const float* y_pred = (const float*)d_in[1]; // [256, 512, 512] float32
  float* out = (float*)d_out;                  // [256, 1] float32

  ctc_alpha_kernel<<<CTC_B, NTHREADS, 0, stream>>>(y_true, y_pred, out);
}